// StaarModelTrainer_326417515372
// MI455X (gfx1250) — compile-verified
//
#include <hip/hip_runtime.h>
#include <hip/hip_bf16.h>

typedef __attribute__((ext_vector_type(2))) float v2f;
typedef __attribute__((ext_vector_type(8))) float v8f;

#define L_SERIES 10239          // 2048 + 8191
#define PAD      80             // zero pad so lag-shifted WMMA reads stay in-bounds
#define NTHREADS 256
#define NWAVES   8              // wave32 -> 256/32
#define SEGLEN   40             // ceil(10239/256)

__constant__ const float kOMEGA = 0.05f;
__constant__ const float kALPHA = 0.10f;
__constant__ const float kBETA  = 0.85f;
// gammaln(2.5) - gammaln(2.0) - 0.5*(log(4) + log(pi))
#define LOG_NORM (-0.98082925301172619f)

__global__ __launch_bounds__(NTHREADS)
void staar_loss_kernel(const float* __restrict__ y, float* __restrict__ out) {
  __shared__ float c[L_SERIES + PAD];
  __shared__ float red[NTHREADS];
  __shared__ float ac[64];
  __shared__ float seg_a[NTHREADS];
  __shared__ float seg_b[NTHREADS];
  __shared__ float seg_in[NTHREADS];
  __shared__ float s_mean, s_var, s_acl;

  const int tid = threadIdx.x;

  // ---- 1) Gather series into LDS: [y[0,:,0], y[1:,-1,0]] ----
  for (int t = tid; t < 2048; t += NTHREADS)
    c[t] = y[(size_t)t * 4];                                   // (0, t, 0)
  for (int i = tid; i < 8191; i += NTHREADS)
    c[2048 + i] = y[(size_t)(i + 1) * 8192 + 8188];            // (i+1, 2047, 0)
  for (int t = L_SERIES + tid; t < L_SERIES + PAD; t += NTHREADS)
    c[t] = 0.0f;
  if (tid < 64) ac[tid] = 0.0f;
  __syncthreads();

  // ---- 2) Mean, then center (padding stays zero) ----
  float p = 0.0f;
  for (int t = tid; t < L_SERIES; t += NTHREADS) p += c[t];
  red[tid] = p; __syncthreads();
  for (int s = NTHREADS / 2; s > 0; s >>= 1) {
    if (tid < s) red[tid] += red[tid + s];
    __syncthreads();
  }
  if (tid == 0) s_mean = red[0] / (float)L_SERIES;
  __syncthreads();
  const float mean = s_mean;
  for (int t = tid; t < L_SERIES; t += NTHREADS) c[t] -= mean;
  __syncthreads();

  // ---- 3) Variance = sum(c^2) ----
  p = 0.0f;
  for (int t = tid; t < L_SERIES; t += NTHREADS) { float v = c[t]; p += v * v; }
  red[tid] = p; __syncthreads();
  for (int s = NTHREADS / 2; s > 0; s >>= 1) {
    if (tid < s) red[tid] += red[tid + s];
    __syncthreads();
  }
  if (tid == 0) s_var = red[0];
  __syncthreads();

  // ---- 4) Autocorrelation lags 0..63 via V_WMMA_F32_16X16X4_F32 ----
  // Per chunk of 4 timesteps t0..t0+3:
  //   B[kk][n] = (n==0) ? c[t0+kk] : 0           (4x16)
  //   A_g[m][kk] = c[t0 + kk + 16*g + m]         (16x4), g = 0..3
  //   D_g[m][0] += sum_kk c[t0+kk]*c[t0+kk+16g+m] -> autocov(lag = 16g+m)
  // A layout: lane<16 -> K={0,1} in {x,y}; lane>=16 -> K={2,3}. m = lane&15.
  // B layout: VGPR x -> rows K=0(K=2), VGPR y -> rows K=1(K=3); column = lane&15.
  {
    const int wave = tid >> 5;
    const int lane = tid & 31;
    const int m    = lane & 15;
    const int half = lane >> 4;          // selects K pair {0,1} vs {2,3}
    // Column mask instead of divergent loads: keeps the hot loop branch-free
    // (uniform LDS broadcast load + VALU multiply, no exec-mask juggling).
    const float bmask = (m == 0) ? 1.0f : 0.0f;
    v8f acc0 = {}, acc1 = {}, acc2 = {}, acc3 = {};
    const int nchunks = (L_SERIES + 3) / 4;   // 2560
    for (int chunk = wave; chunk < nchunks; chunk += NWAVES) {
      const int t0 = chunk * 4;
      v2f b;
      b.x = bmask * c[t0 + 2 * half    ];
      b.y = bmask * c[t0 + 2 * half + 1];
      const int base = t0 + 2 * half + m;
      v2f a;
      a.x = c[base +  0]; a.y = c[base +  1];
      acc0 = __builtin_amdgcn_wmma_f32_16x16x4_f32(false, a, false, b, (short)0, acc0, false, false);
      a.x = c[base + 16]; a.y = c[base + 17];
      acc1 = __builtin_amdgcn_wmma_f32_16x16x4_f32(false, a, false, b, (short)0, acc1, false, false);
      a.x = c[base + 32]; a.y = c[base + 33];
      acc2 = __builtin_amdgcn_wmma_f32_16x16x4_f32(false, a, false, b, (short)0, acc2, false, false);
      a.x = c[base + 48]; a.y = c[base + 49];
      acc3 = __builtin_amdgcn_wmma_f32_16x16x4_f32(false, a, false, b, (short)0, acc3, false, false);
    }
    // Column N=0 lives in lane 0 (rows j) and lane 16 (rows j+8) of each tile.
    if (m == 0) {
      const int rbase = half * 8;
      for (int j = 0; j < 8; ++j) {
        atomicAdd(&ac[ 0 + rbase + j], acc0[j]);
        atomicAdd(&ac[16 + rbase + j], acc1[j]);
        atomicAdd(&ac[32 + rbase + j], acc2[j]);
        atomicAdd(&ac[48 + rbase + j], acc3[j]);
      }
    }
  }
  __syncthreads();

  // ---- 5) ACL loss = mean_{k=1..50} (ac[k]/var)^2 ----
  if (tid == 0) {
    const float var = s_var;
    float sum = 0.0f;
    for (int k = 1; k <= 50; ++k) {
      float a = ac[k] / var;
      sum += a * a;
    }
    s_acl = sum / 50.0f;
  }

  // ---- 6) GARCH NLL via affine segmented scan ----
  // Step: s' = omega + alpha*r2 + beta*s  (affine in s) -> per-segment (a,b).
  const int start = tid * SEGLEN;
  const int end   = (start + SEGLEN < L_SERIES) ? (start + SEGLEN) : L_SERIES;
  float aseg = 1.0f, bseg = 0.0f;
  for (int t = start; t < end; ++t) {
    const float r2 = c[t] * c[t];
    bseg = kOMEGA + kALPHA * r2 + kBETA * bseg;
    aseg *= kBETA;
  }
  seg_a[tid] = aseg; seg_b[tid] = bseg;
  __syncthreads();
  if (tid == 0) {
    float s = kOMEGA / (1.0f - kALPHA - kBETA);   // unconditional variance
    for (int i = 0; i < NTHREADS; ++i) {
      seg_in[i] = s;
      s = seg_a[i] * s + seg_b[i];
    }
  }
  __syncthreads();
  // Replay segment with known incoming state; accumulate -log_prob.
  // scale^2 = s*(NU-2)/NU = 0.5*s ; z2 = r2/(2*s) ; 0.5*(NU+1) = 2.5
  float s = seg_in[tid];
  float nll = 0.0f;
  for (int t = start; t < end; ++t) {
    const float res = c[t];
    const float r2  = res * res;
    s = kOMEGA + kALPHA * r2 + kBETA * s;         // cond_var[t]
    const float z2 = r2 / (2.0f * s);
    nll -= LOG_NORM - 0.5f * logf(0.5f * s) - 2.5f * log1pf(z2);
  }
  red[tid] = nll; __syncthreads();
  for (int sred = NTHREADS / 2; sred > 0; sred >>= 1) {
    if (tid < sred) red[tid] += red[tid + sred];
    __syncthreads();
  }

  // ---- 7) Outputs: (total, acl, garch) ----
  if (tid == 0) {
    const float garch = red[0];
    out[0] = s_acl + garch;
    out[1] = s_acl;
    out[2] = garch;
  }
}

extern "C" void kernel_launch(void* const* d_in, const int* in_sizes, int n_in,
                              void* d_out, int out_size, void* d_ws, size_t ws_size,
                              hipStream_t stream) {
  (void)in_sizes; (void)n_in; (void)d_ws; (void)ws_size; (void)out_size;
  const float* y = (const float*)d_in[0];
  float* out = (float*)d_out;
  // Entire workload touches only ~41 KB of the input; one WGP does it all.
  staar_loss_kernel<<<1, NTHREADS, 0, stream>>>(y, out);
}